// AttentionLayer_18683107737860
// MI455X (gfx1250) — compile-verified
//
#include <hip/hip_runtime.h>
#include <hip/hip_bf16.h>

typedef float v2f __attribute__((ext_vector_type(2)));
typedef float v8f __attribute__((ext_vector_type(8)));

#define USER_DIM 32
#define ITEM_DIM 32
#define IN_DIM   64
#define HID      32

// 16 edges per wave. Full-precision f32 WMMA:
//   D(16x16) = A(16 edges x 4 feats) * B(4 feats x 16 neurons), K-loop c=0..15,
//   two neuron tiles (HID=32).
// Layouts (CDNA5 ISA 7.12.2, 32-bit A 16x4 / B 4x16 / C,D 16x16):
//   A: lane(row=lane&15, half=lane>>4) holds feat[row][4c + 2*half + {0,1}]
//   B: lane holds W1[col + 16t][4c + 2*half + {0,1}],  col = lane&15
//   D: VGPR i holds edge (i + 8*half), neuron col (+16t)
__global__ __launch_bounds__(256) void edge_mlp_wmma_f32(
    const float* __restrict__ uemb,   // [N_USERS, 32]
    const float* __restrict__ iemb,   // [N_ITEMS, 32]
    const int*   __restrict__ eidx,   // [2, E]
    const float* __restrict__ W1,     // [32, 64] row-major
    const float* __restrict__ b1,     // [32]
    const float* __restrict__ W2,     // [32]
    const float* __restrict__ b2,     // [1]
    float* __restrict__ out,          // [E]
    int E)
{
    const int lane = threadIdx.x & 31;
    const int half = lane >> 4;
    const int col  = lane & 15;
    const int koff = 2 * half;

    // ---- Preload W1 B-fragments into registers (amortized over many tiles) ----
    v2f w1r[2][16];
#pragma unroll
    for (int t = 0; t < 2; ++t) {
#pragma unroll
        for (int c = 0; c < 16; ++c) {
            w1r[t][c] = *(const v2f*)(W1 + (col + 16 * t) * IN_DIM + 4 * c + koff);
        }
    }
    const float b1_lo = b1[col];
    const float b1_hi = b1[col + 16];
    const float w2_lo = W2[col];
    const float w2_hi = W2[col + 16];
    const float bias2 = b2[0];

    const int ntiles = (E + 15) >> 4;
    const int wid = (int)((blockIdx.x * blockDim.x + threadIdx.x) >> 5);
    const int nw  = (int)((gridDim.x * blockDim.x) >> 5);

    for (int tile = wid; tile < ntiles; tile += nw) {
        const int ebase = tile << 4;
        int e = ebase + col;
        if (e >= E) e = E - 1;              // clamp; stores predicated below
        const int u  = eidx[e];
        const int it = eidx[E + e];

        // ---- Gather A fragments: one 8B load per lane per K-chunk (L2-resident) ----
        const float* urow = uemb + (long)u  * USER_DIM;
        const float* irow = iemb + (long)it * ITEM_DIM;
        v2f a[16];
#pragma unroll
        for (int c = 0; c < 8; ++c)
            a[c] = *(const v2f*)(urow + 4 * c + koff);
#pragma unroll
        for (int c = 0; c < 8; ++c)
            a[8 + c] = *(const v2f*)(irow + 4 * c + koff);

        // ---- Layer 1: 32x V_WMMA_F32_16X16X4_F32 (full f32 precision) ----
        v8f acc0 = {};
        v8f acc1 = {};
#pragma unroll
        for (int c = 0; c < 16; ++c) {
            acc0 = __builtin_amdgcn_wmma_f32_16x16x4_f32(
                false, a[c], false, w1r[0][c], (short)0, acc0, false, false);
            acc1 = __builtin_amdgcn_wmma_f32_16x16x4_f32(
                false, a[c], false, w1r[1][c], (short)0, acc1, false, false);
        }

        // ---- Bias + LeakyReLU + layer 2 + 16-lane butterfly reduction ----
        float res = 0.f;
#pragma unroll
        for (int i = 0; i < 8; ++i) {
            float h0 = acc0[i] + b1_lo; h0 = (h0 > 0.f) ? h0 : 0.2f * h0;
            float h1 = acc1[i] + b1_hi; h1 = (h1 > 0.f) ? h1 : 0.2f * h1;
            float s = h0 * w2_lo + h1 * w2_hi;
            s += __shfl_xor(s, 1, 32);
            s += __shfl_xor(s, 2, 32);
            s += __shfl_xor(s, 4, 32);
            s += __shfl_xor(s, 8, 32);      // all lanes of each half hold edge (i+8*half)
            if ((lane & 7) == i) res = s;   // lane l now holds edge (l&7)+8*(l>>4)
        }
        // Remap so lanes 0..15 hold edges 0..15, then one coalesced 64B store.
        const int src = (lane < 8) ? lane : (lane + 8);
        res = __shfl(res, src, 32);
        const float logit = res + bias2;
        const float sig = 1.f / (1.f + __expf(-logit));
        const int eo = ebase + lane;
        if (lane < 16 && eo < E) out[eo] = sig;
    }
}

extern "C" void kernel_launch(void* const* d_in, const int* in_sizes, int n_in,
                              void* d_out, int out_size, void* d_ws, size_t ws_size,
                              hipStream_t stream) {
    const float* uemb = (const float*)d_in[0];
    const float* iemb = (const float*)d_in[1];
    const int*   eidx = (const int*)  d_in[2];
    const float* W1   = (const float*)d_in[3];
    const float* b1   = (const float*)d_in[4];
    const float* W2   = (const float*)d_in[5];
    const float* b2   = (const float*)d_in[6];
    float* out = (float*)d_out;

    const int E = out_size;                 // one output per edge
    const int ntiles = (E + 15) / 16;       // 16 edges per wave
    int blocks = (ntiles + 7) / 8;          // 8 waves per 256-thread block
    if (blocks > 2048) blocks = 2048;       // grid-stride; ~12 tiles/wave at E=3.2M
    if (blocks < 1) blocks = 1;

    edge_mlp_wmma_f32<<<blocks, 256, 0, stream>>>(
        uemb, iemb, eidx, W1, b1, W2, b2, out, E);
}